// Block_53128745452102
// MI455X (gfx1250) — compile-verified
//
#include <hip/hip_runtime.h>
#include <hip/hip_bf16.h>

// ---------------------------------------------------------------------------
// MHLA + MoE transformer block for MI455X (gfx1250), fp32 throughout.
// B=8, S=128, E=192, H=6, HD=32, L=64, N_EXP=8, TOP_K=2, ED=768
// ---------------------------------------------------------------------------

#define Bx 8
#define Sx 128
#define Ex 192
#define Hx 6
#define HDx 32
#define Lx 64
#define NEXP 8
#define EDx 768
#define Tx (Bx * Sx)          // 1024 tokens
#define LATK (Sx * Ex)        // 24576
#define LATN (Lx * Ex)        // 12288
#define EPSx 1e-5f

typedef float v2f __attribute__((ext_vector_type(2)));
typedef float v8f __attribute__((ext_vector_type(8)));

// ---------------------------------------------------------------------------
// RMSNorm over last dim (E=192). grid = tokens, block = 192.
// ---------------------------------------------------------------------------
__global__ void rmsnorm_kernel(const float* __restrict__ x,
                               const float* __restrict__ w,
                               float* __restrict__ out) {
  __shared__ float red[Ex];
  const int t = blockIdx.x;
  const int e = threadIdx.x;
  const float v = x[(long long)t * Ex + e];
  red[e] = v * v;
  __syncthreads();
  for (int s = 96; s >= 3; s >>= 1) {
    if (e < s) red[e] += red[e + s];
    __syncthreads();
  }
  if (e == 0) red[0] = red[0] + red[1] + red[2];
  __syncthreads();
  const float scale = rsqrtf(red[0] * (1.0f / Ex) + EPSx);
  out[(long long)t * Ex + e] = v * scale * w[e];
}

// ---------------------------------------------------------------------------
// Latent projection: [8, 24576] @ [24576, 12288], done on the WMMA pipe so the
// 1.21 GB latW stream is bandwidth-bound, not VALU-bound. A is a 16-row matrix
// with rows 0..7 = the 8 batch rows of rmsnorm(x) and rows 8..15 = zero
// (staged in LDS so every lane executes the WMMA with full EXEC).
// Each wave owns a 16-column strip; one V_WMMA_F32_16X16X4_F32 consumes a
// 4x16 (256 B) latW fragment. K is split 4 ways (grid.y) into deterministic
// partial sums. Output rows 0..7 live in VGPRs 0..7 of lanes 0..15.
// grid = (12288/128, 4), block = 256 (8 waves x 16 cols = 128 cols/block).
// ---------------------------------------------------------------------------
__global__ __launch_bounds__(256) void latent_wmma_kernel(
    const float* __restrict__ xn, const float* __restrict__ latW,
    float* __restrict__ part) {
  __shared__ float xs[16][132];  // rows 8..15 zeroed; stride 132 -> no bank conflicts

  const int lane = threadIdx.x & 31;
  const int wave = threadIdx.x >> 5;                 // 0..7
  const int colBase = blockIdx.x * 128 + wave * 16;  // this wave's N strip
  const int seg = blockIdx.y;                        // 0..3
  const int k0seg = seg * (LATK / 4);                // 6144 per segment
  const int lm = lane & 15;
  const int kh = (lane >> 4) * 2;                    // 0 or 2

  v8f c;
#pragma unroll
  for (int r = 0; r < 8; ++r) c[r] = 0.f;

  for (int k0 = 0; k0 < LATK / 4; k0 += 128) {
    // stage A chunk: 16 rows x 128 k (rows 8..15 = 0)
    for (int i = threadIdx.x; i < 16 * 128; i += 256) {
      const int r = i >> 7;
      const int kk = i & 127;
      xs[r][kk] = (r < 8) ? xn[(long long)r * LATK + k0seg + k0 + kk] : 0.f;
    }
    __syncthreads();

    const float* wp = latW + (long long)(k0seg + k0) * LATN + colBase + lm;
    // hint the next chunk of the weight stream into cache
    __builtin_prefetch(wp + (long long)128 * LATN, 0, 1);

#pragma unroll 8
    for (int kk = 0; kk < 128; kk += 4) {
      v2f a, b;
      a.x = xs[lm][kk + kh];
      a.y = xs[lm][kk + kh + 1];
      const float* wq = wp + (long long)(kk + kh) * LATN;
      b.x = wq[0];
      b.y = wq[LATN];
      c = __builtin_amdgcn_wmma_f32_16x16x4_f32(false, a, false, b, (short)0,
                                                c, false, false);
    }
    __syncthreads();
  }

  // rows 0..7 (the batches) sit in c[0..7] of lanes 0..15
  if (lane < 16) {
#pragma unroll
    for (int r = 0; r < 8; ++r)
      part[((long long)seg * Bx + r) * LATN + colBase + lm] = c[r];
  }
}

__global__ void latent_reduce_kernel(const float* __restrict__ part,
                                     const float* __restrict__ latb,
                                     float* __restrict__ latent) {
  const int j = blockIdx.x * 256 + threadIdx.x;  // 0 .. 8*12288-1
  const int n = j % LATN;
  float s = latb[n];
#pragma unroll
  for (int seg = 0; seg < 4; ++seg) s += part[(long long)seg * (Bx * LATN) + j];
  latent[j] = s;
}

// ---------------------------------------------------------------------------
// Generic fp32 WMMA GEMM: C = A[M,K] @ W[K,N] (+bias[N]) (+resid) (ReLU opt),
// optionally batched over grid.z (per-expert weights).
// Block = 256 threads = 8 waves covering a 32(M) x 64(N) tile; each wave does
// one 16x16 tile via V_WMMA_F32_16X16X4_F32, K-loop step 16 with the A tile
// staged in LDS (shared by the 4 waves on the same M-half).
// Requires M%32==0, N%64==0, K%16==0 (true for all call sites).
// ---------------------------------------------------------------------------
__global__ __launch_bounds__(256) void wmma_gemm_kernel(
    const float* __restrict__ A, const float* __restrict__ W,
    const float* __restrict__ bias, const float* __restrict__ resid,
    float* __restrict__ C, int M, int N, int K,
    long long aStride, long long wStride, long long biasStride,
    long long cStride, int relu) {
  __shared__ float As[32][17];  // padded to dodge bank conflicts

  const int z = blockIdx.z;
  A += (long long)z * aStride;
  W += (long long)z * wStride;
  C += (long long)z * cStride;
  const float* bp = bias ? bias + (long long)z * biasStride : nullptr;

  const int lane = threadIdx.x & 31;
  const int wave = threadIdx.x >> 5;  // 0..7
  const int wm = wave >> 2;           // 0..1  (M half)
  const int wn = wave & 3;            // 0..3  (N quarter)
  const int rowBase = blockIdx.y * 32;
  const int tileN = blockIdx.x * 64 + wn * 16;
  const int lm = lane & 15;
  const int kh = (lane >> 4) * 2;     // 0 or 2 : K offset for this half-wave
  const int aRow = wm * 16 + lm;

  const float bv = bp ? bp[tileN + lm] : 0.f;
  v8f c;
#pragma unroll
  for (int r = 0; r < 8; ++r) c[r] = bv;

  for (int k0 = 0; k0 < K; k0 += 16) {
    // stage A[32][16] tile
    for (int i = threadIdx.x; i < 512; i += 256) {
      const int r = i >> 4;
      const int cc = i & 15;
      As[r][cc] = A[(long long)(rowBase + r) * K + k0 + cc];
    }
    __syncthreads();
#pragma unroll
    for (int kk = 0; kk < 16; kk += 4) {
      v2f a, b;
      a.x = As[aRow][kk + kh];
      a.y = As[aRow][kk + kh + 1];
      const float* wp = W + (long long)(k0 + kk + kh) * N + tileN + lm;
      b.x = wp[0];
      b.y = wp[N];
      c = __builtin_amdgcn_wmma_f32_16x16x4_f32(false, a, false, b, (short)0,
                                                c, false, false);
    }
    __syncthreads();
  }

  // C/D layout: VGPR r -> M = r (lanes 0-15) or 8+r (lanes 16-31), N = lane%16
  const int mbase = rowBase + wm * 16 + ((lane >> 4) ? 8 : 0);
  const int n = tileN + lm;
#pragma unroll
  for (int r = 0; r < 8; ++r) {
    float v = c[r];
    const int m = mbase + r;
    if (resid) v += resid[(long long)m * N + n];
    if (relu) v = fmaxf(v, 0.f);
    C[(long long)m * N + n] = v;
  }
}

// ---------------------------------------------------------------------------
// Attention per (b,h): scores = rope(q) @ k^T / sqrt(HD); softmax; @ v.
// grid = B*H = 48, block = 128 (one thread per query row).
// K/V head-slices staged in LDS; online softmax keeps registers small.
// ---------------------------------------------------------------------------
__global__ __launch_bounds__(128) void attn_kernel(
    const float* __restrict__ qb_, const float* __restrict__ kb_,
    const float* __restrict__ vb_, float* __restrict__ ob_) {
  const int bh = blockIdx.x;
  const int b = bh / Hx;
  const int h = bh % Hx;
  __shared__ float ks[Lx][HDx];
  __shared__ float vs[Lx][HDx];
  for (int i = threadIdx.x; i < Lx * HDx; i += 128) {
    const int l = i >> 5;
    const int d = i & 31;
    const long long base = ((long long)b * Lx + l) * Ex + h * HDx + d;
    ks[l][d] = kb_[base];
    vs[l][d] = vb_[base];
  }
  __syncthreads();

  const int s = threadIdx.x;  // query position 0..127
  float q[HDx];
  const float* qp = qb_ + ((long long)b * Sx + s) * Ex + h * HDx;
#pragma unroll
  for (int d = 0; d < HDx; ++d) q[d] = qp[d];

  // RoPE on interleaved pairs (query only, matching reference)
#pragma unroll
  for (int i = 0; i < HDx / 2; ++i) {
    const float inv = powf(10000.f, -(float)(2 * i) / (float)HDx);
    const float ang = (float)s * inv;
    const float cs = cosf(ang), sn = sinf(ang);
    const float x1 = q[2 * i], x2 = q[2 * i + 1];
    q[2 * i] = x1 * cs - x2 * sn;
    q[2 * i + 1] = x1 * sn + x2 * cs;
  }

  const float scale = 0.17677669529663687f;  // 1/sqrt(32)
  float mmax = -1e30f, sum = 0.f;
  float out[HDx];
#pragma unroll
  for (int d = 0; d < HDx; ++d) out[d] = 0.f;

  for (int l = 0; l < Lx; ++l) {
    float sc = 0.f;
#pragma unroll
    for (int d = 0; d < HDx; ++d) sc += q[d] * ks[l][d];
    sc *= scale;
    if (sc > mmax) {
      const float f = expf(mmax - sc);
      sum *= f;
#pragma unroll
      for (int d = 0; d < HDx; ++d) out[d] *= f;
      mmax = sc;
    }
    const float p = expf(sc - mmax);
    sum += p;
#pragma unroll
    for (int d = 0; d < HDx; ++d) out[d] += p * vs[l][d];
  }
  const float invs = 1.f / sum;
  float* op = ob_ + ((long long)b * Sx + s) * Ex + h * HDx;
#pragma unroll
  for (int d = 0; d < HDx; ++d) op[d] = out[d] * invs;
}

// ---------------------------------------------------------------------------
// Router: logits = xn2 @ rW + rb; softmax over 8; top-2 (ties -> lower index,
// matching jax.lax.top_k). grid = 4, block = 256 (one thread per token).
// ---------------------------------------------------------------------------
__global__ __launch_bounds__(256) void router_kernel(
    const float* __restrict__ xn2, const float* __restrict__ rW,
    const float* __restrict__ rb, int* __restrict__ idx,
    float* __restrict__ wout) {
  __shared__ float rws[Ex * NEXP];
  __shared__ float rbs[NEXP];
  for (int i = threadIdx.x; i < Ex * NEXP; i += 256) rws[i] = rW[i];
  if (threadIdx.x < NEXP) rbs[threadIdx.x] = rb[threadIdx.x];
  __syncthreads();

  const int t = blockIdx.x * 256 + threadIdx.x;
  float lg[NEXP];
#pragma unroll
  for (int j = 0; j < NEXP; ++j) lg[j] = rbs[j];
  const float* xp = xn2 + (long long)t * Ex;
  for (int e = 0; e < Ex; ++e) {
    const float xv = xp[e];
#pragma unroll
    for (int j = 0; j < NEXP; ++j) lg[j] += xv * rws[e * NEXP + j];
  }
  float mx = lg[0];
#pragma unroll
  for (int j = 1; j < NEXP; ++j) mx = fmaxf(mx, lg[j]);
  float sm = 0.f;
#pragma unroll
  for (int j = 0; j < NEXP; ++j) {
    lg[j] = expf(lg[j] - mx);
    sm += lg[j];
  }
  const float invs = 1.f / sm;
#pragma unroll
  for (int j = 0; j < NEXP; ++j) lg[j] *= invs;

  int i0 = 0;
  float p0 = lg[0];
#pragma unroll
  for (int j = 1; j < NEXP; ++j)
    if (lg[j] > p0) { p0 = lg[j]; i0 = j; }
  int i1 = -1;
  float p1 = -1.f;
#pragma unroll
  for (int j = 0; j < NEXP; ++j) {
    if (j == i0) continue;
    if (lg[j] > p1) { p1 = lg[j]; i1 = j; }
  }
  idx[t * 2] = i0;
  idx[t * 2 + 1] = i1;
  wout[t * 2] = p0;
  wout[t * 2 + 1] = p1;
}

// ---------------------------------------------------------------------------
// Final combine: out = x1 + p0*eo[i0] + p1*eo[i1]. grid = 1024, block = 192.
// eo layout: [expert][token][E]
// ---------------------------------------------------------------------------
__global__ void combine_kernel(const float* __restrict__ x1,
                               const float* __restrict__ eo,
                               const int* __restrict__ idx,
                               const float* __restrict__ w,
                               float* __restrict__ out) {
  const int t = blockIdx.x;
  const int e = threadIdx.x;
  const int i0 = idx[t * 2];
  const int i1 = idx[t * 2 + 1];
  const float w0 = w[t * 2];
  const float w1 = w[t * 2 + 1];
  const long long o = (long long)t * Ex + e;
  const long long es = (long long)Tx * Ex;
  out[o] = x1[o] + w0 * eo[(long long)i0 * es + o] + w1 * eo[(long long)i1 * es + o];
}

// ---------------------------------------------------------------------------
// Host-side orchestration
// ---------------------------------------------------------------------------
extern "C" void kernel_launch(void* const* d_in, const int* in_sizes, int n_in,
                              void* d_out, int out_size, void* d_ws,
                              size_t ws_size, hipStream_t stream) {
  const float* x    = (const float*)d_in[0];
  const float* rms1 = (const float*)d_in[1];
  const float* rms2 = (const float*)d_in[2];
  const float* latW = (const float*)d_in[3];
  const float* latb = (const float*)d_in[4];
  const float* qW   = (const float*)d_in[5];
  const float* qb   = (const float*)d_in[6];
  const float* kW   = (const float*)d_in[7];
  const float* kb   = (const float*)d_in[8];
  const float* vW   = (const float*)d_in[9];
  const float* vb   = (const float*)d_in[10];
  const float* oW   = (const float*)d_in[11];
  const float* ob   = (const float*)d_in[12];
  const float* rW   = (const float*)d_in[13];
  const float* rb   = (const float*)d_in[14];
  const float* e1W  = (const float*)d_in[15];
  const float* e1b  = (const float*)d_in[16];
  const float* swW  = (const float*)d_in[17];
  const float* swb  = (const float*)d_in[18];
  const float* e2W  = (const float*)d_in[19];
  const float* e2b  = (const float*)d_in[20];

  float* ws = (float*)d_ws;
  size_t off = 0;
  float* xn1    = ws + off; off += (size_t)Tx * Ex;          // 196608
  float* latent = ws + off; off += (size_t)Bx * Lx * Ex;     // 98304
  float* qbuf   = ws + off; off += (size_t)Tx * Ex;
  float* kbuf   = ws + off; off += (size_t)Bx * Lx * Ex;
  float* vbuf   = ws + off; off += (size_t)Bx * Lx * Ex;
  float* attnb  = ws + off; off += (size_t)Tx * Ex;
  float* x1     = ws + off; off += (size_t)Tx * Ex;
  float* xn2    = ws + off; off += (size_t)Tx * Ex;
  float* part   = ws + off; off += (size_t)4 * Bx * LATN;    // 393216
  float* t1     = ws + off; off += (size_t)NEXP * Tx * EDx;  // 6291456
  float* t2     = ws + off; off += (size_t)NEXP * Tx * EDx;
  float* eo     = ws + off; off += (size_t)NEXP * Tx * Ex;   // 1572864
  float* rw_w   = ws + off; off += (size_t)Tx * 2;
  int*   rw_idx = (int*)(ws + off); off += (size_t)Tx * 2;

  // ---- 1. xn1 = rmsnorm(x, rms1_w)
  rmsnorm_kernel<<<Tx, Ex, 0, stream>>>(x, rms1, xn1);

  // ---- 2-3. latent = xn1_flat @ latW + latb  (WMMA-fed bandwidth stream)
  latent_wmma_kernel<<<dim3(LATN / 128, 4), 256, 0, stream>>>(xn1, latW, part);
  latent_reduce_kernel<<<(Bx * LATN) / 256, 256, 0, stream>>>(part, latb, latent);

  // ---- 4-6. Q / K / V projections (WMMA)
  wmma_gemm_kernel<<<dim3(Ex / 64, Tx / 32, 1), 256, 0, stream>>>(
      xn1, qW, qb, nullptr, qbuf, Tx, Ex, Ex, 0, 0, 0, 0, 0);
  wmma_gemm_kernel<<<dim3(Ex / 64, (Bx * Lx) / 32, 1), 256, 0, stream>>>(
      latent, kW, kb, nullptr, kbuf, Bx * Lx, Ex, Ex, 0, 0, 0, 0, 0);
  wmma_gemm_kernel<<<dim3(Ex / 64, (Bx * Lx) / 32, 1), 256, 0, stream>>>(
      latent, vW, vb, nullptr, vbuf, Bx * Lx, Ex, Ex, 0, 0, 0, 0, 0);

  // ---- 7. attention (RoPE + online softmax)
  attn_kernel<<<Bx * Hx, Sx, 0, stream>>>(qbuf, kbuf, vbuf, attnb);

  // ---- 8. output projection + residual: x1 = x + attn @ oW + ob
  wmma_gemm_kernel<<<dim3(Ex / 64, Tx / 32, 1), 256, 0, stream>>>(
      attnb, oW, ob, x, x1, Tx, Ex, Ex, 0, 0, 0, 0, 0);

  // ---- 9. xn2 = rmsnorm(x1, rms2_w)
  rmsnorm_kernel<<<Tx, Ex, 0, stream>>>(x1, rms2, xn2);

  // ---- 10. router softmax + top-2
  router_kernel<<<Tx / 256, 256, 0, stream>>>(xn2, rW, rb, rw_idx, rw_w);

  // ---- 11-13. dense expert GEMMs, batched over grid.z = 8 experts (WMMA)
  wmma_gemm_kernel<<<dim3(EDx / 64, Tx / 32, NEXP), 256, 0, stream>>>(
      xn2, e1W, e1b, nullptr, t1, Tx, EDx, Ex,
      0, (long long)Ex * EDx, EDx, (long long)Tx * EDx, 0);
  wmma_gemm_kernel<<<dim3(EDx / 64, Tx / 32, NEXP), 256, 0, stream>>>(
      t1, swW, swb, nullptr, t2, Tx, EDx, EDx,
      (long long)Tx * EDx, (long long)EDx * EDx, EDx, (long long)Tx * EDx, 1);
  wmma_gemm_kernel<<<dim3(Ex / 64, Tx / 32, NEXP), 256, 0, stream>>>(
      t2, e2W, e2b, nullptr, eo, Tx, Ex, EDx,
      (long long)Tx * EDx, (long long)EDx * Ex, Ex, (long long)Tx * Ex, 0);

  // ---- 14. out = x1 + sum_k pk * eo[idx_k]
  combine_kernel<<<Tx, Ex, 0, stream>>>(x1, eo, rw_idx, rw_w, (float*)d_out);
}